// Conv2DTransposeNN_14018773254483
// MI455X (gfx1250) — compile-verified
//
#include <hip/hip_runtime.h>

typedef float v2f __attribute__((ext_vector_type(2)));
typedef float v4f __attribute__((ext_vector_type(4)));
typedef float v8f __attribute__((ext_vector_type(8)));

// Problem shape (fixed by the reference): NHWC f32
constexpr int B_ = 32, H_ = 64, W_ = 64, C_ = 64;   // kernel_size = (2,2)
constexpr int LDS_STRIDE = 68;                      // 64 + 4 pad -> conflict-free WMMA reads

__global__ __launch_bounds__(128)
void conv2dtranspose_nn_kernel(const float* __restrict__ in, float* __restrict__ out)
{
    __shared__ float lds_in[W_ * LDS_STRIDE];       // 64*68*4 = 17408 B (of 320 KB/WGP)
    __shared__ float s_sum[W_];                     // per-pixel channel sums * C

    const int tid = threadIdx.x;
    const int blk = blockIdx.x;                     // b*H + h
    const int b   = blk / H_;
    const int h   = blk % H_;

    const float* row_in = in + (size_t)(b * H_ + h) * (W_ * C_);

    // ---- Phase 1: coalesced b128 load of one input row (64 px x 64 ch = 16 KB) into padded LDS
    #pragma unroll
    for (int j = 0; j < (W_ * C_ / 4) / 128; ++j) {          // 1024 float4 / 128 thr = 8 iters
        int li = j * 128 + tid;                              // float4 index within row
        int p  = li >> 4;                                    // pixel (16 float4 per pixel)
        int c0 = (li & 15) << 2;                             // channel offset
        v4f v = *(const v4f*)(row_in + (size_t)li * 4);
        *(v4f*)(&lds_in[p * LDS_STRIDE + c0]) = v;
    }
    __syncthreads();

    // ---- Phase 2: channel reduction via V_WMMA_F32_16X16X4_F32 (exact f32 math).
    // Each of the 4 waves reduces 16 pixels; B = ones(4x16) so D[m][n] = sum_k A[m][k].
    const int wave = tid >> 5;
    const int lane = tid & 31;
    const int m    = lane & 15;                              // A-matrix row M within tile
    const int half = lane >> 4;                              // lanes 0-15: K={0,1}; 16-31: K={2,3}
    const int p    = wave * 16 + m;                          // absolute input pixel

    v8f acc = {};
    v2f ones; ones.x = 1.0f; ones.y = 1.0f;
    #pragma unroll
    for (int k = 0; k < C_ / 4; ++k) {                       // 16 wmma steps cover K = 64 channels
        v2f a = *(const v2f*)(&lds_in[p * LDS_STRIDE + k * 4 + half * 2]);
        acc = __builtin_amdgcn_wmma_f32_16x16x4_f32(
                  /*neg_a=*/false, a, /*neg_b=*/false, ones,
                  /*c_mod=*/(short)0, acc, /*reuse_a=*/false, /*reuse_b=*/false);
    }

    // C/D layout: lane group half, VGPR r -> M = r + 8*half (same value in every column).
    // Lanes 0 and 16 of each wave publish the 16 row sums, pre-scaled by C (=64).
    if (m == 0) {
        #pragma unroll
        for (int r = 0; r < 8; ++r)
            s_sum[wave * 16 + half * 8 + r] = acc[r] * (float)C_;
    }
    __syncthreads();

    // ---- Phase 3: broadcast-store two output rows (y = 2h, 2h+1), 16384 floats contiguous.
    // Each value repeats over 128 consecutive floats (2 x-cells * 64 channels).
    float* row_out = out + ((size_t)b * (H_ * 2) + (size_t)(h * 2)) * (size_t)(W_ * 2 * C_);
    #pragma unroll
    for (int j = 0; j < (2 * W_ * 2 * C_ / 4) / 128; ++j) {  // 4096 float4 / 128 thr = 32 iters
        int i4  = j * 128 + tid;
        int src = (i4 & 2047) >> 5;                          // (float4 idx % 2048) / 32 -> pixel
        float v = s_sum[src];
        v4f o; o.x = v; o.y = v; o.z = v; o.w = v;
        *(v4f*)(row_out + (size_t)i4 * 4) = o;
    }
}

extern "C" void kernel_launch(void* const* d_in, const int* in_sizes, int n_in,
                              void* d_out, int out_size, void* d_ws, size_t ws_size,
                              hipStream_t stream) {
    (void)in_sizes; (void)n_in; (void)out_size; (void)d_ws; (void)ws_size;
    const float* in = (const float*)d_in[0];
    float* out = (float*)d_out;
    // One block per input row (b,h): 32*64 = 2048 blocks, 128 threads (4 waves) each.
    conv2dtranspose_nn_kernel<<<B_ * H_, 128, 0, stream>>>(in, out);
}